// SocialInfluenceModule_50818053046585
// MI455X (gfx1250) — compile-verified
//
#include <hip/hip_runtime.h>

#define NNODES 50000
#define NEDGES 1600000
#define DIM 128
#define NEG_SLOPE 0.2f

typedef __attribute__((ext_vector_type(2))) float v2f;
typedef __attribute__((ext_vector_type(8))) float v8f;

// -------------------------------------------------------------------------
// Edge scatter: 128 lanes per edge, one feature per lane (coalesced gather),
// hardware f32 atomics into agg[dst], lane 0 counts the in-degree.
// -------------------------------------------------------------------------
__global__ __launch_bounds__(256)
void scatter_kernel(const float* __restrict__ x,
                    const int* __restrict__ edge_index,   // [2, E] flat: src then dst
                    const float* __restrict__ ew,
                    float* __restrict__ agg,
                    float* __restrict__ cnt) {
    const long long t = (long long)blockIdx.x * blockDim.x + threadIdx.x;
    const long long e = t >> 7;          // edge id
    const int d = (int)(t & 127);        // feature id
    if (e >= NEDGES) return;
    const int src = edge_index[e];
    const int dst = edge_index[(long long)NEDGES + e];
    const float w = ew[e];
    const float v = x[(long long)src * DIM + d] * w;
    unsafeAtomicAdd(&agg[(long long)dst * DIM + d], v);
    if (d == 0) unsafeAtomicAdd(&cnt[dst], 1.0f);
}

// -------------------------------------------------------------------------
// Fused GCN layer GEMM:
//   out = LeakyReLU( x @ Ws^T + neigh @ Wn^T + (bs + bn + lb) )
//   neigh = (agg + x) / (cnt + 1)   formed in registers
// One wave per 16x16 output tile, V_WMMA_F32_16X16X4_F32, K-loop over 128.
// Block = 256 threads = 8 waves -> 8 N-tiles (all 128 cols) for 16 rows.
// grid.x = 50000/16 = 3125 (exact).
// -------------------------------------------------------------------------
__global__ __launch_bounds__(256, 2)
void gemm_layer_kernel(const float* __restrict__ x,
                       const float* __restrict__ agg,
                       const float* __restrict__ cnt,
                       const float* __restrict__ Ws,
                       const float* __restrict__ Wn,
                       const float* __restrict__ bs,
                       const float* __restrict__ bn,
                       const float* __restrict__ lb,
                       float* __restrict__ out) {
    const int lane  = threadIdx.x & 31;
    const int ntile = threadIdx.x >> 5;          // 0..7
    const int mtile = blockIdx.x;                // 0..3124
    const int lhalf = lane >> 4;                 // 0: K={0,1}, 1: K={2,3}
    const int lmod  = lane & 15;
    const int m = mtile * 16 + lmod;             // A-fragment row for this lane
    const int n = ntile * 16 + lmod;             // B-fragment column for this lane

    const float rc = 1.0f / (cnt[m] + 1.0f);

    const float* __restrict__ xrow  = x   + (long long)m * DIM;
    const float* __restrict__ arow  = agg + (long long)m * DIM;
    const float* __restrict__ wsrow = Ws  + (long long)n * DIM;  // B[k][n] = Ws[n][k]
    const float* __restrict__ wnrow = Wn  + (long long)n * DIM;

    v8f c = {};
#pragma unroll 4
    for (int k0 = 0; k0 < DIM; k0 += 4) {
        const int kk = k0 + 2 * lhalf;
        v2f ax  = *(const v2f*)(xrow + kk);
        v2f aa  = *(const v2f*)(arow + kk);
        v2f an  = (ax + aa) * rc;                // neigh operand on the fly
        v2f bsv = *(const v2f*)(wsrow + kk);
        v2f bnv = *(const v2f*)(wnrow + kk);
        c = __builtin_amdgcn_wmma_f32_16x16x4_f32(false, ax, false, bsv,
                                                  (short)0, c, false, false);
        c = __builtin_amdgcn_wmma_f32_16x16x4_f32(false, an, false, bnv,
                                                  (short)0, c, false, false);
    }

    const float bsum = bs[n] + bn[n] + lb[n];
#pragma unroll
    for (int r = 0; r < 8; ++r) {
        const int row = mtile * 16 + r + 8 * lhalf;   // C/D layout: VGPR r -> M = r (+8)
        float v = c[r] + bsum;
        v = (v > 0.0f) ? v : NEG_SLOPE * v;
        out[(long long)row * DIM + n] = v;
    }
}

// -------------------------------------------------------------------------
// Final projection: out = x @ W^T + b   (no activation)
// -------------------------------------------------------------------------
__global__ __launch_bounds__(256, 2)
void gemm_agg_kernel(const float* __restrict__ x,
                     const float* __restrict__ W,
                     const float* __restrict__ b,
                     float* __restrict__ out) {
    const int lane  = threadIdx.x & 31;
    const int ntile = threadIdx.x >> 5;
    const int mtile = blockIdx.x;
    const int lhalf = lane >> 4;
    const int lmod  = lane & 15;
    const int m = mtile * 16 + lmod;
    const int n = ntile * 16 + lmod;

    const float* __restrict__ xrow = x + (long long)m * DIM;
    const float* __restrict__ wrow = W + (long long)n * DIM;

    v8f c = {};
#pragma unroll 4
    for (int k0 = 0; k0 < DIM; k0 += 4) {
        const int kk = k0 + 2 * lhalf;
        v2f av = *(const v2f*)(xrow + kk);
        v2f bv = *(const v2f*)(wrow + kk);
        c = __builtin_amdgcn_wmma_f32_16x16x4_f32(false, av, false, bv,
                                                  (short)0, c, false, false);
    }

    const float bias = b[n];
#pragma unroll
    for (int r = 0; r < 8; ++r) {
        const int row = mtile * 16 + r + 8 * lhalf;
        out[(long long)row * DIM + n] = c[r] + bias;
    }
}

// -------------------------------------------------------------------------
extern "C" void kernel_launch(void* const* d_in, const int* in_sizes, int n_in,
                              void* d_out, int out_size, void* d_ws, size_t ws_size,
                              hipStream_t stream) {
    const float* x0     = (const float*)d_in[0];   // user_embeddings [N,D]
    const int*   ei     = (const int*)  d_in[1];   // edge_index [2,E]
    const float* ew     = (const float*)d_in[2];   // edge weights [E]
    const float* Wself  = (const float*)d_in[3];   // [2,D,D]
    const float* bself  = (const float*)d_in[4];   // [2,D]
    const float* Wneigh = (const float*)d_in[5];   // [2,D,D]
    const float* bneigh = (const float*)d_in[6];   // [2,D]
    const float* lbias  = (const float*)d_in[7];   // [2,D]
    const float* Wagg   = (const float*)d_in[8];   // [D,D]
    const float* bagg   = (const float*)d_in[9];   // [D]
    float* out = (float*)d_out;

    // workspace layout (floats): agg[N*D] | cnt[N] | xb0[N*D] | xb1[N*D]
    float* agg = (float*)d_ws;
    float* cnt = agg + (size_t)NNODES * DIM;
    float* xb0 = cnt + NNODES;
    float* xb1 = xb0 + (size_t)NNODES * DIM;

    const long long scatter_threads = (long long)NEDGES * DIM;  // 204.8M
    const int scatter_blocks = (int)((scatter_threads + 255) / 256);
    const int gemm_blocks = NNODES / 16;                        // 3125, exact

    for (int l = 0; l < 2; ++l) {
        const float* xin = (l == 0) ? x0 : xb0;
        float* xout      = (l == 0) ? xb0 : xb1;

        hipMemsetAsync(agg, 0, (size_t)NNODES * DIM * sizeof(float), stream);
        hipMemsetAsync(cnt, 0, (size_t)NNODES * sizeof(float), stream);

        scatter_kernel<<<scatter_blocks, 256, 0, stream>>>(xin, ei, ew, agg, cnt);

        gemm_layer_kernel<<<gemm_blocks, 256, 0, stream>>>(
            xin, agg, cnt,
            Wself  + (size_t)l * DIM * DIM,
            Wneigh + (size_t)l * DIM * DIM,
            bself + l * DIM, bneigh + l * DIM, lbias + l * DIM,
            xout);
    }

    gemm_agg_kernel<<<gemm_blocks, 256, 0, stream>>>(xb1, Wagg, bagg, out);
}